// DGN_14877766713834
// MI455X (gfx1250) — compile-verified
//
#include <hip/hip_runtime.h>

// ---------------------------------------------------------------------------
// NNConv x3 + CBT for MI455X (gfx1250, wave32)
//   N=512 nodes, E=N*N edges, V=4 edge feats, layers (1->36, 36->24, 24->8)
// Strategy: fuse edge-MLP (WMMA f32 16x16x4, K==V==4) + ReLU + per-edge
// matvec + LDS scatter-accumulate per 16-edge tile. Never materialize [E,864].
// ---------------------------------------------------------------------------

#define NN   512
#define EE   (NN * NN)
#define VV   4

typedef float v2f __attribute__((ext_vector_type(2)));
typedef float v8f __attribute__((ext_vector_type(8)));

// ---------------------------------------------------------------------------
__global__ void zero_u32_kernel(unsigned* __restrict__ p, int n) {
    int i = blockIdx.x * blockDim.x + threadIdx.x;
    if (i < n) p[i] = 0u;
}

__global__ void count_kernel(const long long* __restrict__ eidx,
                             int* __restrict__ cnt) {
    int e = blockIdx.x * blockDim.x + threadIdx.x;
    if (e < EE) atomicAdd(&cnt[(int)eidx[EE + e]], 1);
}

// ---------------------------------------------------------------------------
// Fused NNConv edge kernel.
//  CIN/COUT: layer dims. CTP: cin*cout padded to multiple of 16.
//  WSTRIDE: CTP+4 so the (row+8) lane-half store lands 32 banks away.
// Block: 256 threads = 8 waves; each wave owns column-tiles ct = wave, wave+8...
// ---------------------------------------------------------------------------
template <int CIN, int COUT, int CTP, int WSTRIDE>
__global__ void nnconv_edge_kernel(const float* __restrict__ x,        // [N,CIN]
                                   const float* __restrict__ ea,       // [E,4]
                                   const long long* __restrict__ eidx, // [2,E]
                                   const float* __restrict__ lin_w,    // [4,CIN*COUT]
                                   const float* __restrict__ lin_b,    // [CIN*COUT]
                                   float* __restrict__ agg)            // [N,COUT]
{
    extern __shared__ float smem[];
    float* lw_s  = smem;                  // 4*CTP   (zero-padded lin_w)
    float* b_s   = lw_s + 4 * CTP;        // CTP     (zero-padded bias)
    float* x_s   = b_s + CTP;             // N*CIN   (all node features)
    float* agg_s = x_s + NN * CIN;        // N*COUT  (block-local accumulator)
    float* w_s   = agg_s + NN * COUT;     // 16*WSTRIDE (per-tile edge weights)
    int*   src_s = (int*)(w_s + 16 * WSTRIDE);
    int*   dst_s = src_s + 16;

    const int tid = threadIdx.x;

    for (int c = tid; c < CTP; c += blockDim.x) {
        const bool ok = (c < CIN * COUT);
#pragma unroll
        for (int k = 0; k < 4; ++k)
            lw_s[k * CTP + c] = ok ? lin_w[k * (CIN * COUT) + c] : 0.0f;
        b_s[c] = ok ? lin_b[c] : 0.0f;
    }
    for (int i = tid; i < NN * CIN;  i += blockDim.x) x_s[i]   = x[i];
    for (int i = tid; i < NN * COUT; i += blockDim.x) agg_s[i] = 0.0f;
    __syncthreads();

    const int lane = tid & 31;
    const int wave = tid >> 5;
    const int half = lane >> 4;   // K-half selector (ISA 16x4 f32 A layout)
    const int l16  = lane & 15;
    constexpr int NT = CTP / 16;  // column tiles of the edge-MLP output

    for (int t = blockIdx.x; t < EE / 16; t += gridDim.x) {
        const int te = t * 16;
        if (tid < 16) {
            src_s[tid] = (int)eidx[te + tid];
            dst_s[tid] = (int)eidx[EE + te + tid];
        }

        // A operand: 16 edges x K=4. Lane L<16 holds K={0,1} of edge L,
        // lane L>=16 holds K={2,3} of edge L-16 (2 consecutive floats).
        v2f a;
        {
            const float2 av =
                *(const float2*)(ea + (size_t)(te + l16) * VV + half * 2);
            a.x = av.x; a.y = av.y;
        }

        for (int ct = wave; ct < NT; ct += 8) {
            const int col = ct * 16 + l16;
            v2f b;                       // B: 4x16 slice of lin_w, mirrored layout
            b.x = lw_s[(half * 2 + 0) * CTP + col];
            b.y = lw_s[(half * 2 + 1) * CTP + col];
            const float bv = b_s[col];   // bias broadcast down the M dim
            v8f c = {bv, bv, bv, bv, bv, bv, bv, bv};

            v8f d = __builtin_amdgcn_wmma_f32_16x16x4_f32(
                false, a, false, b, (short)0, c, false, false);

            // ReLU + stage tile: VGPR r holds row (r + half*8), col = ct*16+l16
#pragma unroll
            for (int r = 0; r < 8; ++r) {
                float v = d[r] > 0.0f ? d[r] : 0.0f;
                w_s[(r + half * 8) * WSTRIDE + col] = v;
            }
        }
        __syncthreads();

        // Per-edge matvec: msg[e][o] = sum_i x[src[e]][i] * w[e][i*COUT+o]
        for (int idx = tid; idx < 16 * COUT; idx += blockDim.x) {
            const int e = idx / COUT;
            const int o = idx - e * COUT;
            const int s = src_s[e];
            float acc = 0.0f;
#pragma unroll
            for (int i = 0; i < CIN; ++i)
                acc = fmaf(x_s[s * CIN + i], w_s[e * WSTRIDE + i * COUT + o], acc);
            atomicAdd(&agg_s[dst_s[e] * COUT + o], acc);   // ds_add_f32
        }
        __syncthreads();
    }

    // flush block-local accumulator -> global (native f32 atomic add)
    for (int i = tid; i < NN * COUT; i += blockDim.x) {
        float v = agg_s[i];
        if (v != 0.0f) unsafeAtomicAdd(&agg[i], v);
    }
}

// ---------------------------------------------------------------------------
// h_out = relu(agg / max(cnt,1) + h_prev @ root + bias)
// ---------------------------------------------------------------------------
template <int CIN, int COUT>
__global__ void finalize_kernel(const float* __restrict__ agg,
                                const int* __restrict__ cnt,
                                const float* __restrict__ hprev,
                                const float* __restrict__ root,  // [CIN,COUT]
                                const float* __restrict__ bias,  // [COUT]
                                float* __restrict__ hout) {
    int idx = blockIdx.x * blockDim.x + threadIdx.x;
    if (idx >= NN * COUT) return;
    const int n = idx / COUT;
    const int o = idx - n * COUT;
    const int cc = cnt[n];
    float v = agg[idx] / (float)(cc > 1 ? cc : 1) + bias[o];
#pragma unroll
    for (int i = 0; i < CIN; ++i)
        v = fmaf(hprev[n * CIN + i], root[i * COUT + o], v);
    hout[idx] = fmaxf(v, 0.0f);
}

// ---------------------------------------------------------------------------
// CBT: out[i,j] = sum_d |h[j,d] - h[i,d]| , h: [N,8]
// ---------------------------------------------------------------------------
__global__ void cbt_kernel(const float* __restrict__ h, float* __restrict__ out) {
    __shared__ float hs[NN * 8];
    for (int i = threadIdx.x; i < NN * 8; i += blockDim.x) hs[i] = h[i];
    __syncthreads();
    int p = blockIdx.x * blockDim.x + threadIdx.x;
    if (p >= NN * NN) return;
    const int i = p >> 9;
    const int j = p & (NN - 1);
    float s = 0.0f;
#pragma unroll
    for (int d = 0; d < 8; ++d) s += fabsf(hs[j * 8 + d] - hs[i * 8 + d]);
    out[p] = s;
}

// ---------------------------------------------------------------------------
extern "C" void kernel_launch(void* const* d_in, const int* in_sizes, int n_in,
                              void* d_out, int out_size, void* d_ws, size_t ws_size,
                              hipStream_t stream) {
    const float*     x    = (const float*)d_in[0];
    const float*     ea   = (const float*)d_in[1];
    const long long* eidx = (const long long*)d_in[2];
    const float *lw1 = (const float*)d_in[3],  *lb1 = (const float*)d_in[4];
    const float *rt1 = (const float*)d_in[5],  *bs1 = (const float*)d_in[6];
    const float *lw2 = (const float*)d_in[7],  *lb2 = (const float*)d_in[8];
    const float *rt2 = (const float*)d_in[9],  *bs2 = (const float*)d_in[10];
    const float *lw3 = (const float*)d_in[11], *lb3 = (const float*)d_in[12];
    const float *rt3 = (const float*)d_in[13], *bs3 = (const float*)d_in[14];

    // workspace layout (floats)
    float* ws = (float*)d_ws;
    float* h1  = ws;                 // 512*36
    float* h2  = h1 + NN * 36;       // 512*24
    float* h3  = h2 + NN * 24;       // 512*8
    float* agg = h3 + NN * 8;        // 512*36 (max cout)
    int*   cnt = (int*)(agg + NN * 36); // 512

    // dynamic LDS sizes per layer
    auto smem_bytes = [](int cin, int cout, int ctp, int wstride) -> size_t {
        return (size_t)(4 * ctp + ctp + NN * cin + NN * cout + 16 * wstride) * 4
               + 2 * 16 * sizeof(int);
    };
    const size_t sm1 = smem_bytes(1, 36, 48, 52);
    const size_t sm2 = smem_bytes(36, 24, 864, 868);
    const size_t sm3 = smem_bytes(24, 8, 192, 196);
    (void)hipFuncSetAttribute((const void*)nnconv_edge_kernel<1, 36, 48, 52>,
                              hipFuncAttributeMaxDynamicSharedMemorySize, (int)sm1);
    (void)hipFuncSetAttribute((const void*)nnconv_edge_kernel<36, 24, 864, 868>,
                              hipFuncAttributeMaxDynamicSharedMemorySize, (int)sm2);
    (void)hipFuncSetAttribute((const void*)nnconv_edge_kernel<24, 8, 192, 196>,
                              hipFuncAttributeMaxDynamicSharedMemorySize, (int)sm3);

    const int TB = 256;
    const int NB = 256;  // edge-kernel blocks; fewer blocks -> fewer atomic flushes

    // counts (fixed for all layers)
    zero_u32_kernel<<<(NN + TB - 1) / TB, TB, 0, stream>>>((unsigned*)cnt, NN);
    count_kernel<<<(EE + TB - 1) / TB, TB, 0, stream>>>(eidx, cnt);

    // ---- layer 1: 1 -> 36 ----
    zero_u32_kernel<<<(NN * 36 + TB - 1) / TB, TB, 0, stream>>>((unsigned*)agg, NN * 36);
    nnconv_edge_kernel<1, 36, 48, 52><<<NB, TB, sm1, stream>>>(x, ea, eidx, lw1, lb1, agg);
    finalize_kernel<1, 36><<<(NN * 36 + TB - 1) / TB, TB, 0, stream>>>(agg, cnt, x, rt1, bs1, h1);

    // ---- layer 2: 36 -> 24 ----
    zero_u32_kernel<<<(NN * 24 + TB - 1) / TB, TB, 0, stream>>>((unsigned*)agg, NN * 24);
    nnconv_edge_kernel<36, 24, 864, 868><<<NB, TB, sm2, stream>>>(h1, ea, eidx, lw2, lb2, agg);
    finalize_kernel<36, 24><<<(NN * 24 + TB - 1) / TB, TB, 0, stream>>>(agg, cnt, h1, rt2, bs2, h2);

    // ---- layer 3: 24 -> 8 ----
    zero_u32_kernel<<<(NN * 8 + TB - 1) / TB, TB, 0, stream>>>((unsigned*)agg, NN * 8);
    nnconv_edge_kernel<24, 8, 192, 196><<<NB, TB, sm3, stream>>>(h2, ea, eidx, lw3, lb3, agg);
    finalize_kernel<24, 8><<<(NN * 8 + TB - 1) / TB, TB, 0, stream>>>(agg, cnt, h2, rt3, bs3, h3);

    // ---- CBT ----
    cbt_kernel<<<(NN * NN + TB - 1) / TB, TB, 0, stream>>>(h3, (float*)d_out);
}